// Mamba_52201032516095
// MI455X (gfx1250) — compile-verified
//
#include <hip/hip_runtime.h>
#include <hip/hip_bf16.h>

typedef __attribute__((ext_vector_type(2))) float v2f;
typedef __attribute__((ext_vector_type(8))) float v8f;

#define EPSF 1e-5f

// ---------------------------------------------------------------- stats
__global__ void k_stats(const float* __restrict__ x, float* __restrict__ meanv,
                        float* __restrict__ stdv) {
  __shared__ float s1[256], s2[256];
  int bm = blockIdx.x, tid = threadIdx.x;
  const float* row = x + (size_t)bm * 2048;
  float s = 0.f, ss = 0.f;
  for (int i = tid; i < 2048; i += 256) { float v = row[i]; s += v; ss += v * v; }
  s1[tid] = s; s2[tid] = ss; __syncthreads();
  for (int o = 128; o > 0; o >>= 1) {
    if (tid < o) { s1[tid] += s1[tid + o]; s2[tid] += s2[tid + o]; }
    __syncthreads();
  }
  if (tid == 0) {
    float m = s1[0] * (1.f / 2048.f);
    float var = s2[0] * (1.f / 2048.f) - m * m;
    meanv[bm] = m;
    stdv[bm] = sqrtf(var + EPSF);
  }
}

// ---------------------------------------------------------------- patch embed (WMMA)
// u[token,128] = patches[token,16] @ W_pe[16,128] + b_pe + pos[n]
__global__ void k_patch(const float* __restrict__ x, const float* __restrict__ Wpe,
                        const float* __restrict__ bpe, const float* __restrict__ pos,
                        const float* __restrict__ meanv, const float* __restrict__ stdv,
                        float* __restrict__ u) {
  int lane = threadIdx.x & 31;
  int wave = blockIdx.x * 8 + (threadIdx.x >> 5);   // 4096 tiles
  int tm = wave >> 3, tn = wave & 7;
  int hi = lane >> 4, lr = lane & 15;
  int t0 = tm * 16;
  int bm = t0 >> 8, n0 = t0 & 255;                  // 16 | 256, never crosses bm
  float m = meanv[bm], rs = 1.f / stdv[bm];
  const float* xr = x + (size_t)bm * 2048;
  int nrow = n0 + lr;
  int col = tn * 16 + lr;
  v8f acc = {0.f,0.f,0.f,0.f,0.f,0.f,0.f,0.f};
  #pragma unroll
  for (int k0 = 0; k0 < 16; k0 += 4) {
    int kk = k0 + 2 * hi;
    int i0 = nrow * 8 + kk;     if (i0 > 2047) i0 = 2047;   // repeat-last padding
    int i1 = nrow * 8 + kk + 1; if (i1 > 2047) i1 = 2047;
    v2f a, b;
    a.x = (xr[i0] - m) * rs;
    a.y = (xr[i1] - m) * rs;
    b.x = Wpe[kk * 128 + col];
    b.y = Wpe[(kk + 1) * 128 + col];
    acc = __builtin_amdgcn_wmma_f32_16x16x4_f32(false, a, false, b, (short)0, acc,
                                                false, false);
  }
  #pragma unroll
  for (int r = 0; r < 8; r++) {
    int row = r + 8 * hi;
    int token = t0 + row, n = n0 + row;
    u[(size_t)token * 128 + col] = acc[r] + bpe[col] + pos[n * 128 + col];
  }
}

// ---------------------------------------------------------------- generic WMMA GEMM
// C[T,N] = A[T,K] @ B[K,N] (+resid). lda==K (multiple of 16). Column guard for N.
__global__ void k_gemm(const float* __restrict__ A, const float* __restrict__ B,
                       const float* __restrict__ resid, float* __restrict__ C,
                       int K, int Nvalid, int tiles_n, int ldb, int ldc) {
  int lane = threadIdx.x & 31;
  int wave = blockIdx.x * (blockDim.x >> 5) + (threadIdx.x >> 5);
  int tm = wave / tiles_n, tn = wave % tiles_n;
  int hi = lane >> 4, lr = lane & 15;
  int col = tn * 16 + lr;
  bool cok = col < Nvalid;
  const float* Arow = A + (size_t)(tm * 16 + lr) * K + 2 * hi;
  const float* Bc = B + (size_t)(2 * hi) * ldb + (cok ? col : 0);
  v8f acc = {0.f,0.f,0.f,0.f,0.f,0.f,0.f,0.f};
  for (int k = 0; k < K; k += 16) {                   // 4 WMMAs per trip
    #pragma unroll
    for (int kk = 0; kk < 16; kk += 4) {
      v2f a, b;
      a.x = Arow[k + kk];
      a.y = Arow[k + kk + 1];
      const float* bp = Bc + (size_t)(k + kk) * ldb;
      b.x = cok ? bp[0] : 0.f;
      b.y = cok ? bp[ldb] : 0.f;
      acc = __builtin_amdgcn_wmma_f32_16x16x4_f32(false, a, false, b, (short)0, acc,
                                                  false, false);
    }
  }
  if (cok) {
    #pragma unroll
    for (int r = 0; r < 8; r++) {
      int row = tm * 16 + r + 8 * hi;
      size_t idx = (size_t)row * ldc + col;
      float v = acc[r];
      if (resid) v += resid[idx];
      C[idx] = v;
    }
  }
}

// ---------------------------------------------------------------- per-layer RMSNorm
__global__ void k_rmsnorm(const float* __restrict__ u, const float* __restrict__ w,
                          float* __restrict__ o) {
  __shared__ float s1[128];
  int row = blockIdx.x, tid = threadIdx.x;
  float v = u[(size_t)row * 128 + tid];
  s1[tid] = v * v; __syncthreads();
  for (int o2 = 64; o2 > 0; o2 >>= 1) {
    if (tid < o2) s1[tid] += s1[tid + o2];
    __syncthreads();
  }
  float sc = rsqrtf(s1[0] * (1.f / 128.f) + EPSF);
  o[(size_t)row * 128 + tid] = v * sc * w[tid];
}

// ---------------------------------------------------------------- causal dwconv + SiLU
__global__ void k_conv(const float* __restrict__ xz, const float* __restrict__ cw,
                       const float* __restrict__ cb, float* __restrict__ xs) {
  int token = blockIdx.x, d = threadIdx.x;
  int t = token & 255;
  float acc = cb[d];
  #pragma unroll
  for (int j = 0; j < 4; j++) {
    int tt = t - 3 + j;
    if (tt >= 0) acc += xz[(size_t)(token - 3 + j) * 512 + d] * cw[d * 4 + j];
  }
  float sig = 1.f / (1.f + __expf(-acc));
  xs[(size_t)token * 256 + d] = acc * sig;
}

// ---------------------------------------------------------------- fused SSM scan
// one block per sequence, one thread per channel; B/C/dt staged via LDS
__global__ void k_scan(const float* __restrict__ dbc, const float* __restrict__ xs,
                       const float* __restrict__ xz, const float* __restrict__ Wd,
                       const float* __restrict__ bd, const float* __restrict__ Al,
                       const float* __restrict__ Dp, float* __restrict__ y) {
  __shared__ float sB[16], sC[16], sD8[8];
  int bm = blockIdx.x, d = threadIdx.x;
  float An[16], h[16], wdc[8];
  #pragma unroll
  for (int s = 0; s < 16; s++) { An[s] = -__expf(Al[d * 16 + s]); h[s] = 0.f; }
  #pragma unroll
  for (int r = 0; r < 8; r++) wdc[r] = Wd[r * 256 + d];
  float bdd = bd[d], dpd = Dp[d];
  for (int t = 0; t < 256; t++) {
    int token = bm * 256 + t;
    if (d < 16)       sB[d]       = dbc[(size_t)token * 48 + 8 + d];
    else if (d < 32)  sC[d - 16]  = dbc[(size_t)token * 48 + 8 + d];   // 24+(d-16)
    else if (d < 40)  sD8[d - 32] = dbc[(size_t)token * 48 + d - 32];
    __syncthreads();
    float xv = xs[(size_t)token * 256 + d];
    float dl = bdd;
    #pragma unroll
    for (int r = 0; r < 8; r++) dl += sD8[r] * wdc[r];
    dl = (dl > 20.f) ? dl : logf(1.f + __expf(dl));     // softplus
    float acc = 0.f;
    #pragma unroll
    for (int s = 0; s < 16; s++) {
      h[s] = __expf(dl * An[s]) * h[s] + dl * sB[s] * xv;
      acc += h[s] * sC[s];
    }
    acc += dpd * xv;
    float zv = xz[(size_t)token * 512 + 256 + d];
    y[(size_t)token * 256 + d] = acc * (zv / (1.f + __expf(-zv)));
    __syncthreads();
  }
}

// ---------------------------------------------------------------- final RMS + LayerNorm
__global__ void k_finalnorm(const float* __restrict__ u, const float* __restrict__ rw,
                            const float* __restrict__ g, const float* __restrict__ b,
                            float* __restrict__ o) {
  __shared__ float s1[128], s2[128];
  int row = blockIdx.x, tid = threadIdx.x;
  float v = u[(size_t)row * 128 + tid];
  s1[tid] = v * v; __syncthreads();
  for (int o2 = 64; o2 > 0; o2 >>= 1) {
    if (tid < o2) s1[tid] += s1[tid + o2];
    __syncthreads();
  }
  float t = v * rsqrtf(s1[0] * (1.f / 128.f) + EPSF) * rw[tid];
  __syncthreads();
  s1[tid] = t; s2[tid] = t * t; __syncthreads();
  for (int o2 = 64; o2 > 0; o2 >>= 1) {
    if (tid < o2) { s1[tid] += s1[tid + o2]; s2[tid] += s2[tid + o2]; }
    __syncthreads();
  }
  float mu = s1[0] * (1.f / 128.f);
  float var = s2[0] * (1.f / 128.f) - mu * mu;
  o[(size_t)row * 128 + tid] = (t - mu) * rsqrtf(var + EPSF) * g[tid] + b[tid];
}

// ---------------------------------------------------------------- head GEMM (WMMA) + denorm
// out[32,96] = (A[32,32768] @ Wh[32768,96] + bh) * std + mean
__global__ void k_head(const float* __restrict__ A, const float* __restrict__ Wh,
                       const float* __restrict__ bh, const float* __restrict__ meanv,
                       const float* __restrict__ stdv, float* __restrict__ out) {
  int lane = threadIdx.x & 31;
  int wave = threadIdx.x >> 5;            // 12 waves, one block
  int tm = wave / 6, tn = wave % 6;
  int hi = lane >> 4, lr = lane & 15;
  int col = tn * 16 + lr;
  const float* Arow = A + (size_t)(tm * 16 + lr) * 32768 + 2 * hi;
  const float* Bc = Wh + (size_t)(2 * hi) * 96 + col;
  v8f acc = {0.f,0.f,0.f,0.f,0.f,0.f,0.f,0.f};
  for (int k = 0; k < 32768; k += 32) {               // 8 WMMAs per trip
    if (k + 288 <= 32768) {                           // stream-ahead prefetch
      __builtin_prefetch(Arow + k + 256, 0, 1);       // -> global_prefetch_b8
      __builtin_prefetch(Bc + (size_t)(k + 256) * 96, 0, 1);
    }
    #pragma unroll
    for (int kk = 0; kk < 32; kk += 4) {
      v2f a, b;
      a.x = Arow[k + kk];
      a.y = Arow[k + kk + 1];
      b.x = Bc[(size_t)(k + kk) * 96];
      b.y = Bc[(size_t)(k + kk + 1) * 96];
      acc = __builtin_amdgcn_wmma_f32_16x16x4_f32(false, a, false, b, (short)0, acc,
                                                  false, false);
    }
  }
  #pragma unroll
  for (int r = 0; r < 8; r++) {
    int row = tm * 16 + r + 8 * hi;
    out[row * 96 + col] = (acc[r] + bh[col]) * stdv[row] + meanv[row];
  }
}

// ================================================================ launch
extern "C" void kernel_launch(void* const* d_in, const int* in_sizes, int n_in,
                              void* d_out, int out_size, void* d_ws, size_t ws_size,
                              hipStream_t stream) {
  const float* x      = (const float*)d_in[0];
  const float* W_pe   = (const float*)d_in[1];
  const float* b_pe   = (const float*)d_in[2];
  const float* pos    = (const float*)d_in[3];
  const float* rms_w  = (const float*)d_in[4];
  const float* W_in   = (const float*)d_in[5];
  const float* conv_w = (const float*)d_in[6];
  const float* conv_b = (const float*)d_in[7];
  const float* W_xp   = (const float*)d_in[8];
  const float* W_dt   = (const float*)d_in[9];
  const float* b_dt   = (const float*)d_in[10];
  const float* A_log  = (const float*)d_in[11];
  const float* D_ssm  = (const float*)d_in[12];
  const float* W_out  = (const float*)d_in[13];
  const float* rmsf_w = (const float*)d_in[14];
  const float* ln_g   = (const float*)d_in[15];
  const float* ln_b   = (const float*)d_in[16];
  const float* W_head = (const float*)d_in[17];
  const float* b_head = (const float*)d_in[18];
  float* out = (float*)d_out;

  float* ws = (float*)d_ws;
  float* w_mean = ws;                     // 32
  float* w_std  = ws + 32;                // 32
  float* w_u    = ws + 64;                // 8192*128
  float* w_un   = w_u  + 1048576;         // 8192*128
  float* w_xz   = w_un + 1048576;         // 8192*512
  float* w_xs   = w_xz + 4194304;         // 8192*256
  float* w_dbc  = w_xs + 2097152;         // 8192*48
  float* w_y    = w_dbc + 393216;         // 8192*256

  k_stats<<<32, 256, 0, stream>>>(x, w_mean, w_std);
  k_patch<<<512, 256, 0, stream>>>(x, W_pe, b_pe, pos, w_mean, w_std, w_u);

  for (int l = 0; l < 2; l++) {
    k_rmsnorm<<<8192, 128, 0, stream>>>(w_u, rms_w + l * 128, w_un);
    // xz = un @ W_in : 8192x128x512 -> 512*32 tiles
    k_gemm<<<2048, 256, 0, stream>>>(w_un, W_in + (size_t)l * 128 * 512, nullptr,
                                     w_xz, 128, 512, 32, 512, 512);
    k_conv<<<8192, 256, 0, stream>>>(w_xz, conv_w + l * 256 * 4, conv_b + l * 256, w_xs);
    // dbc = xs @ W_xp : 8192x256x40 (pad N to 48) -> 512*3 tiles
    k_gemm<<<192, 256, 0, stream>>>(w_xs, W_xp + (size_t)l * 256 * 40, nullptr,
                                    w_dbc, 256, 40, 3, 40, 48);
    k_scan<<<32, 256, 0, stream>>>(w_dbc, w_xs, w_xz, W_dt + l * 8 * 256,
                                   b_dt + l * 256, A_log + l * 256 * 16,
                                   D_ssm + l * 256, w_y);
    // u = y @ W_out + u : 8192x256x128 -> 512*8 tiles
    k_gemm<<<512, 256, 0, stream>>>(w_y, W_out + (size_t)l * 256 * 128, w_u,
                                    w_u, 256, 128, 8, 128, 128);
  }

  k_finalnorm<<<8192, 128, 0, stream>>>(w_u, rmsf_w, ln_g, ln_b, w_un);
  k_head<<<1, 384, 0, stream>>>(w_un, W_head, b_head, w_mean, w_std, out);
}